// SphericalNSS_60868276519530
// MI455X (gfx1250) — compile-verified
//
#include <hip/hip_runtime.h>

// Problem constants (match reference): H,W = 1024,2048 ; B,N = 8,64
#define HH 1024
#define WW 2048
#define BB 8
#define NN 64

typedef __attribute__((ext_vector_type(2))) float v2f;
typedef __attribute__((ext_vector_type(8))) float v8f;

#if defined(__has_builtin)
#if __has_builtin(__builtin_amdgcn_wmma_f32_16x16x4_f32)
#define USE_WMMA 1
#endif
#endif

// One block per (batch b, fixation n). 256 threads = 8 wave32.
// Only the LAST fixation per (b, row) does the work (dedup); it replays the
// (compacted, order-preserved) fixations for that row to resolve overlaps.
__global__ __launch_bounds__(256) void SphericalNSS_rows_kernel(
    const float* __restrict__ y_pred,     // (B,1,H,W) f32
    const float* __restrict__ fixations,  // (B,N,2)   f32
    const float* __restrict__ edge_vals,  // (H)       f32
    const int*   __restrict__ ker_ws,     // (H)       i32
    float*       __restrict__ partials)   // (B*N)     f32 scratch
{
    __shared__ __align__(16) float rowbuf[WW];  // 8 KB staged y_pred row
    __shared__ int   s_xs[NN];
    __shared__ int   s_ys[NN];
    __shared__ int   s_left[NN];              // compacted window lefts
    __shared__ int   s_cnt;
    __shared__ float s_red[256];

    const int bid = blockIdx.x;
    const int b   = bid / NN;
    const int n   = bid % NN;
    const int t   = threadIdx.x;

    // Decode fixations for this batch once into LDS (rintf == round-half-even,
    // matching jnp.rint).
    if (t < NN) {
        float fx = fixations[((size_t)b * NN + t) * 2 + 0];
        float fy = fixations[((size_t)b * NN + t) * 2 + 1];
        s_xs[t] = (int)rintf(fx * (float)(WW - 1));
        s_ys[t] = (int)rintf(fy * (float)(HH - 1));
    }
    __syncthreads();

    const int y = s_ys[n];

    // Representative = last fixation in scan order touching this row.
    bool rep = true;
    for (int m = n + 1; m < NN; ++m)
        if (s_ys[m] == y) { rep = false; break; }
    if (!rep) {                       // uniform across the block
        if (t == 0) partials[bid] = 0.0f;
        return;
    }

    // ---- CDNA5 async global -> LDS staging of the y_pred row (8 KB) ----
    // 2 x (256 lanes x 16B) = 8KB via ASYNCcnt; overlapped with the scalar
    // table loads and fixation compaction below, then waited + barrier.
    const float* row = y_pred + ((size_t)b * HH + (size_t)y) * (size_t)WW;
    {
        unsigned l0 = (unsigned)(uintptr_t)(&rowbuf[t * 4]);
        const float* g0 = row + t * 4;
        asm volatile("global_load_async_to_lds_b128 %0, %1, off"
                     :: "v"(l0), "v"(g0) : "memory");
        unsigned l1 = (unsigned)(uintptr_t)(&rowbuf[(t + 256) * 4]);
        const float* g1 = row + (t + 256) * 4;
        asm volatile("global_load_async_to_lds_b128 %0, %1, off"
                     :: "v"(l1), "v"(g1) : "memory");
    }

    const bool  pole = (y == 0) || (y == HH - 1);
    const int   kw   = ker_ws[y];
    const float ev   = edge_vals[y];

    // Compact the fixations hitting this row (order preserved => the forward
    // replay below keeps last-writer-wins semantics). Usually s_cnt == 1.
    if (t == 0) {
        int k = 0;
        for (int m = 0; m < NN; ++m)
            if (s_ys[m] == y) s_left[k++] = s_xs[m] - (kw >> 1);
        s_cnt = k;
    }

    asm volatile("s_wait_asynccnt 0" ::: "memory");
    __syncthreads();
    const int K = s_cnt;

    const int wave = t >> 5;
    const int lane = t & 31;
    const int cbase = wave * 256 + lane * 2;   // this thread's first column

    // fm values for this thread's 8 columns (4 pairs, 64 cols apart), in regs.
    float vv[8];
    if (pole) {
        #pragma unroll
        for (int i = 0; i < 8; ++i) vv[i] = 1.0f;
    } else {
        #pragma unroll
        for (int i = 0; i < 8; ++i) vv[i] = 0.0f;   // fm starts at zero
        for (int k = 0; k < K; ++k) {               // one DS broadcast per k
            const int left = s_left[k];
            #pragma unroll
            for (int it = 0; it < 4; ++it) {
                const int c0 = cbase + it * 64;
                const int o0 = (c0 - left) & (WW - 1);       // W is pow2
                const int o1 = (c0 + 1 - left) & (WW - 1);
                if (o0 < kw) vv[2 * it]     = (o0 == 0 || o0 == kw - 1) ? ev : 1.0f;
                if (o1 < kw) vv[2 * it + 1] = (o1 == 0 || o1 == kw - 1) ? ev : 1.0f;
            }
        }
    }

#ifdef USE_WMMA
    v8f acc = {};                       // 16x16 f32 accumulator (8 VGPRs)
    const v2f ones = {1.0f, 1.0f};      // A = all-ones 16x4 (layout-invariant)
#else
    float accf = 0.0f;
#endif

    #pragma unroll
    for (int it = 0; it < 4; ++it) {
        const int c0 = cbase + it * 64;
        const float p0 = vv[2 * it]     * rowbuf[c0];
        const float p1 = vv[2 * it + 1] * rowbuf[c0 + 1];
#ifdef USE_WMMA
        // C += ones(16x4) x B(4x16): every C element accumulates a column-sum
        // of B; total wave sum = sum of C row 0 (VGPR0, lanes 0..15).
        v2f bv = {p0, p1};
        acc = __builtin_amdgcn_wmma_f32_16x16x4_f32(
            false, ones, false, bv, (short)0, acc, false, false);
#else
        accf += p0 + p1;
#endif
    }

    // Per-wave result -> LDS, fixed-shape tree sum (deterministic).
#ifdef USE_WMMA
    s_red[t] = (lane < 16) ? acc[0] : 0.0f;   // C row 0 lives in lanes 0..15
#else
    s_red[t] = accf;
#endif
    __syncthreads();
    for (int s = 128; s > 0; s >>= 1) {
        if (t < s) s_red[t] += s_red[t + s];
        __syncthreads();
    }
    if (t == 0) partials[bid] = s_red[0];
}

// Single-block deterministic reduction of the B*N partials.
__global__ __launch_bounds__(256) void SphericalNSS_final_kernel(
    const float* __restrict__ partials, float* __restrict__ out)
{
    __shared__ float red[256];
    const int t = threadIdx.x;
    red[t] = partials[t] + partials[t + 256];   // B*N = 512
    __syncthreads();
    for (int s = 128; s > 0; s >>= 1) {
        if (t < s) red[t] += red[t + s];
        __syncthreads();
    }
    if (t == 0) out[0] = red[0] / (float)(BB * NN);
}

extern "C" void kernel_launch(void* const* d_in, const int* in_sizes, int n_in,
                              void* d_out, int out_size, void* d_ws, size_t ws_size,
                              hipStream_t stream) {
    const float* y_pred    = (const float*)d_in[0];
    const float* fixations = (const float*)d_in[1];
    const float* edge_vals = (const float*)d_in[2];
    const int*   ker_ws    = (const int*)d_in[3];
    float* partials = (float*)d_ws;             // B*N floats = 2 KB scratch

    SphericalNSS_rows_kernel<<<BB * NN, 256, 0, stream>>>(
        y_pred, fixations, edge_vals, ker_ws, partials);
    SphericalNSS_final_kernel<<<1, 256, 0, stream>>>(partials, (float*)d_out);
}